// MultiModeVMD_43130061586618
// MI455X (gfx1250) — compile-verified
//
#include <hip/hip_runtime.h>
#include <stdint.h>

// MultiModeVMD on gfx1250: 4 separable Gaussian blurs as banded f32 WMMA
// (V_WMMA_F32_16X16X4_F32) + mode differencing. Memory-bound target:
// 768 MiB traffic -> ~33us at 23.3 TB/s. Input tile fill uses the CDNA5
// async global->LDS path (ASYNCcnt) instead of VGPR staging.

typedef __attribute__((ext_vector_type(2))) float v2f;
typedef __attribute__((ext_vector_type(8))) float v8f;

#define H_IMG 256
#define W_IMG 256
#define TILE  32
#define HALO  8
#define IN_T  48   // TILE + 2*HALO

__device__ __forceinline__ float bandw(const float* g, int d, int twop) {
    return (d >= 0 && d <= twop) ? g[d] : 0.0f;
}

// Horizontal pass: one wave filters all 48 rows x 32 cols of its blur.
// Out = A(16xKw) x G(Kw x16), G banded from LDS weight table.
template<int SIZE, int PAD>
__device__ __forceinline__ void hpass(const float* __restrict__ sin_,
                                      float* __restrict__ shb,
                                      const float* __restrict__ g,
                                      int lane) {
    constexpr int TWOP = 2 * PAD;
    constexpr int NK   = (16 + TWOP) / 4;
    const int hf = lane >> 4;      // 0 or 1 (lane half)
    const int lc = lane & 15;

    // Banded weight fragments for B (reused across all 6 tiles)
    v2f bf[NK];
#pragma unroll
    for (int kk = 0; kk < NK; ++kk) {
        const int j0 = kk * 4 + (hf << 1);
        bf[kk].x = bandw(g, j0 - lc, TWOP);
        bf[kk].y = bandw(g, j0 + 1 - lc, TWOP);
    }

#pragma unroll
    for (int t = 0; t < 6; ++t) {
        const int r0 = (t >> 1) << 4;              // 0,16,32
        const int c0 = (t & 1) << 4;               // 0,16
        const int m  = r0 + lc;                    // A row
        const int cb = c0 + (HALO - PAD) + (hf << 1);  // even -> 8B aligned
        v8f acc = {0.f, 0.f, 0.f, 0.f, 0.f, 0.f, 0.f, 0.f};
#pragma unroll
        for (int kk = 0; kk < NK; ++kk) {
            v2f a;
            a.x = sin_[m * IN_T + cb + kk * 4];
            a.y = sin_[m * IN_T + cb + kk * 4 + 1];
            acc = __builtin_amdgcn_wmma_f32_16x16x4_f32(
                false, a, false, bf[kk], (short)0, acc, false, false);
        }
        // D layout: VGPR i, lanes<16 -> row i, lanes>=16 -> row i+8; col = lane&15
#pragma unroll
        for (int i = 0; i < 8; ++i) {
            const int rr = r0 + (hf << 3) + i;
            shb[rr * TILE + c0 + lc] = acc[i];
        }
    }
}

// Vertical pass: weights in A (transposed band), H data in B.
template<int SIZE, int PAD>
__device__ __forceinline__ v8f vpass(const float* __restrict__ shb,
                                     const float* __restrict__ g,
                                     int lane, int r0, int c0) {
    constexpr int TWOP = 2 * PAD;
    constexpr int NK   = (16 + TWOP) / 4;
    const int hf = lane >> 4;
    const int lc = lane & 15;
    v8f acc = {0.f, 0.f, 0.f, 0.f, 0.f, 0.f, 0.f, 0.f};
#pragma unroll
    for (int kk = 0; kk < NK; ++kk) {
        const int j0 = kk * 4 + (hf << 1);
        v2f a;                                   // A[i,j] = g[j - i]
        a.x = bandw(g, j0 - lc, TWOP);
        a.y = bandw(g, j0 + 1 - lc, TWOP);
        const int row = r0 + (HALO - PAD) + j0;  // B[j,c] = H[row][c0+c]
        v2f b;
        b.x = shb[row * TILE + c0 + lc];
        b.y = shb[(row + 1) * TILE + c0 + lc];
        acc = __builtin_amdgcn_wmma_f32_16x16x4_f32(
            false, a, false, b, (short)0, acc, false, false);
    }
    return acc;
}

__global__ __launch_bounds__(128)
void vmd_wmma_kernel(const float* __restrict__ x,
                     float* __restrict__ out,
                     int planes) {
    __shared__ float s_in[IN_T * IN_T];       // 48x48 input tile + halo
    __shared__ float s_h[4][IN_T * TILE];     // horizontal result per blur
    __shared__ float s_g[4][20];              // Gaussian weight tables

    const int tid  = threadIdx.x;
    const int lane = tid & 31;
    const int wv   = tid >> 5;

    const int bx    = blockIdx.x;
    const int plane = bx >> 6;
    const int t6    = bx & 63;
    const int gy    = (t6 >> 3) << 5;
    const int gx    = (t6 & 7) << 5;

    // Async tile fill: in-bounds lanes DMA global->LDS (ASYNCcnt path),
    // out-of-bounds lanes write the zero padding directly to LDS.
    // Each s_in element is written exactly once -> no cross-path hazard.
    const float* xp = x + (size_t)plane * (H_IMG * W_IMG);
    for (int idx = tid; idx < IN_T * IN_T; idx += 128) {
        const int r = idx / IN_T, c = idx - r * IN_T;
        const int gr = gy + r - HALO, gc = gx + c - HALO;
        if (gr >= 0 && gr < H_IMG && gc >= 0 && gc < W_IMG) {
            const float* gp = xp + gr * W_IMG + gc;
            const unsigned lds_off = (unsigned)(uintptr_t)(&s_in[idx]);
            asm volatile("global_load_async_to_lds_b32 %0, %1, off"
                         :: "v"(lds_off), "v"(gp)
                         : "memory");
        } else {
            s_in[idx] = 0.f;
        }
    }

    // Per-block Gaussian weight generation (44 expf total; overlaps the DMA)
    if (tid < 4) {
        const int   sizes[4]  = {17, 13, 9, 5};
        const float sigmas[4] = {2.5f, 1.8f, 1.2f, 0.6f};
        const int n = sizes[tid];
        const float s = sigmas[tid];
        float sum = 0.f;
        for (int i = 0; i < n; ++i) {
            const float d = (float)(i - n / 2);
            const float w = __expf(-d * d / (2.f * s * s));
            s_g[tid][i] = w;
            sum += w;
        }
        const float inv = 1.f / sum;
        for (int i = 0; i < n; ++i) s_g[tid][i] *= inv;
    }

    // Drain this wave's async copies, then make them visible block-wide.
    asm volatile("s_wait_asynccnt 0x0" ::: "memory");
    __syncthreads();

    // Horizontal pass: wave w owns blur w (wave-uniform branch, EXEC full)
    if (wv == 0)      hpass<17, 8>(s_in, s_h[0], s_g[0], lane);
    else if (wv == 1) hpass<13, 6>(s_in, s_h[1], s_g[1], lane);
    else if (wv == 2) hpass<9,  4>(s_in, s_h[2], s_g[2], lane);
    else              hpass<5,  2>(s_in, s_h[3], s_g[3], lane);
    __syncthreads();

    // Vertical pass: wave w owns one 16x16 output quadrant, all 4 blurs
    const int r0 = (wv >> 1) << 4;
    const int c0 = (wv & 1) << 4;
    const v8f V0 = vpass<17, 8>(s_h[0], s_g[0], lane, r0, c0);
    const v8f V1 = vpass<13, 6>(s_h[1], s_g[1], lane, r0, c0);
    const v8f V2 = vpass<9,  4>(s_h[2], s_g[2], lane, r0, c0);
    const v8f V3 = vpass<5,  2>(s_h[3], s_g[3], lane, r0, c0);

    // Mode differencing + coalesced stores
    const int hf = lane >> 4, lc = lane & 15;
    const size_t mode_stride = (size_t)planes * (H_IMG * W_IMG);
    const int col = gx + c0 + lc;
    float* op = out + (size_t)plane * (H_IMG * W_IMG) + col;
#pragma unroll
    for (int i = 0; i < 8; ++i) {
        const int rr   = r0 + (hf << 3) + i;      // tile-local row
        const int grow = gy + rr;
        const float xv = s_in[(rr + HALO) * IN_T + (c0 + lc + HALO)];
        const size_t off = (size_t)grow * W_IMG;
        op[off]                   = V0[i];
        op[off + mode_stride]     = V1[i] - V0[i];
        op[off + 2 * mode_stride] = V2[i] - V1[i];
        op[off + 3 * mode_stride] = V3[i] - V2[i];
        op[off + 4 * mode_stride] = xv    - V3[i];
    }
}

extern "C" void kernel_launch(void* const* d_in, const int* in_sizes, int n_in,
                              void* d_out, int out_size, void* d_ws, size_t ws_size,
                              hipStream_t stream) {
    (void)n_in; (void)out_size; (void)d_ws; (void)ws_size;
    const float* x = (const float*)d_in[0];
    float* out = (float*)d_out;
    const int planes = in_sizes[0] / (H_IMG * W_IMG);   // B*C = 512
    const int blocks = planes * 64;                     // 8x8 tiles per plane
    vmd_wmma_kernel<<<dim3(blocks), dim3(128), 0, stream>>>(x, out, planes);
}